// SemanticRefinement_85718957293800
// MI455X (gfx1250) — compile-verified
//
#include <hip/hip_runtime.h>
#include <hip/hip_bf16.h>
#include <math.h>

#define WB_B   8
#define WB_NE  4096
#define WB_NC  128
#define WB_H   512
#define K_DROP 1228              // int(0.3 * 4096)
#define KEEP_N (WB_NE - K_DROP)  // 2868

typedef __attribute__((ext_vector_type(2))) float v2f;
typedef __attribute__((ext_vector_type(4))) float v4f;
typedef __attribute__((ext_vector_type(8))) float v8f;

__device__ __forceinline__ float wave_reduce_sum(float v) {
#pragma unroll
    for (int off = 16; off > 0; off >>= 1) v += __shfl_xor(v, off, 32);
    return v;
}
__device__ __forceinline__ float wave_reduce_max(float v) {
#pragma unroll
    for (int off = 16; off > 0; off >>= 1) v = fmaxf(v, __shfl_xor(v, off, 32));
    return v;
}

// ---------------------------------------------------------------------------
// K0: csum[b*NC + m] = sum_h H_c[b,m,h]   (one wave per row)
// ---------------------------------------------------------------------------
__global__ __launch_bounds__(128) void csum_kernel(const float* __restrict__ Hc,
                                                   float* __restrict__ csum) {
    const int gw   = (blockIdx.x * blockDim.x + threadIdx.x) >> 5;
    const int lane = threadIdx.x & 31;
    if (gw >= WB_B * WB_NC) return;
    const float* p = Hc + (size_t)gw * WB_H;
    float s = 0.f;
    for (int k = lane; k < WB_H; k += 32) s += p[k];
    s = wave_reduce_sum(s);
    if (lane == 0) csum[gw] = s;
}

// ---------------------------------------------------------------------------
// K1: per (b, 16-row tile): logits[16x128] = He_tile @ Hc^T via WMMA f32
//     16x16x4, then fused softmax / relevance / score1 / combine.
//     Block = 256 threads = 8 waves; wave w owns claim columns [16w,16w+16).
// ---------------------------------------------------------------------------
__global__ __launch_bounds__(256) void score_kernel(
    const float* __restrict__ He, const float* __restrict__ Hc,
    const float* __restrict__ Wscore, const float* __restrict__ csum,
    float* __restrict__ score_out)
{
    __shared__ float lds_logits[16][WB_NC + 4];

    const int tid  = threadIdx.x;
    const int lane = tid & 31;
    const int wave = tid >> 5;                 // 0..7
    const int blocks_per_b = WB_NE / 16;
    const int b    = blockIdx.x / blocks_per_b;
    const int row0 = (blockIdx.x % blocks_per_b) * 16;

    const int half = lane >> 4;                // 0 or 1
    const int lm   = lane & 15;

    // A frag: lane (half,lm) holds He[row0+lm, k0+2*half .. +1]
    // B frag: lane (half,lm) holds Hc[16w+lm, k0+2*half .. +1]  (B = Hc^T)
    const float* pA = He + ((size_t)b * WB_NE + row0 + lm) * WB_H + 2 * half;
    const float* pB = Hc + ((size_t)b * WB_NC + wave * 16 + lm) * WB_H + 2 * half;

    v8f acc = {};
#pragma unroll 4
    for (int k0 = 0; k0 < WB_H; k0 += 4) {
        v2f av = *(const v2f*)(pA + k0);
        v2f bv = *(const v2f*)(pB + k0);
        acc = __builtin_amdgcn_wmma_f32_16x16x4_f32(
            /*neg_a=*/false, av, /*neg_b=*/false, bv,
            /*c_mod=*/(short)0, acc, /*reuse_a=*/false, /*reuse_b=*/false);
    }

    // C layout: vgpr v, lanes 0-15 -> (M=v, N=lane), lanes 16-31 -> (M=v+8, N=lane-16)
#pragma unroll
    for (int v = 0; v < 8; ++v)
        lds_logits[v + 8 * half][wave * 16 + lm] = acc[v];
    __syncthreads();

    // wave `wave` handles rows 2*wave and 2*wave+1 of the 16-row tile
#pragma unroll
    for (int rr = 0; rr < 2; ++rr) {
        const int r = wave * 2 + rr;
        float l0 = lds_logits[r][lane];
        float l1 = lds_logits[r][lane + 32];
        float l2 = lds_logits[r][lane + 64];
        float l3 = lds_logits[r][lane + 96];
        float mx = wave_reduce_max(fmaxf(fmaxf(l0, l1), fmaxf(l2, l3)));
        float e0 = expf(l0 - mx), e1 = expf(l1 - mx);
        float e2 = expf(l2 - mx), e3 = expf(l3 - mx);
        const float* cs = csum + b * WB_NC;
        float num = e0 * cs[lane] + e1 * cs[lane + 32] + e2 * cs[lane + 64] + e3 * cs[lane + 96];
        float den = e0 + e1 + e2 + e3;
        num = wave_reduce_sum(num);
        den = wave_reduce_sum(den);
        const float rel = num / den;           // relevance_sum for this node

        // score1 = dot(He[row], Wscore)
        const float* he = He + ((size_t)b * WB_NE + row0 + r) * WB_H;
        float s1 = 0.f;
        for (int k = lane; k < WB_H; k += 32) s1 += he[k] * Wscore[k];
        s1 = wave_reduce_sum(s1);

        if (lane == 0) {
            // ALPHA=0.5: 0.5*score1 + 0.5*(-log(rel+1e-10))
            score_out[(size_t)b * WB_NE + row0 + r] = 0.5f * s1 - 0.5f * logf(rel + 1e-10f);
        }
    }
}

// ---------------------------------------------------------------------------
// K2: one GGNN step for ONE batch (hidden dim 1 => all weights scalar).
//     a = w_msg * (adj @ h) + b_msg ; gated update.
//     Block = 512 threads = 16 waves = 16 rows; h staged in LDS.
//     NT=true on the final step for this batch: non-temporal adj reads
//     (last use -> don't keep lines in L2).
// ---------------------------------------------------------------------------
template <bool NT>
__global__ __launch_bounds__(512) void ggnn_step_kernel(
    const float* __restrict__ adj_b, const float* __restrict__ h_in_b,
    float* __restrict__ h_out_b,
    const float* __restrict__ Wm, const float* __restrict__ bm,
    const float* __restrict__ Wzp, const float* __restrict__ Uzp, const float* __restrict__ bzp,
    const float* __restrict__ Wrp, const float* __restrict__ Urp, const float* __restrict__ brp,
    const float* __restrict__ Whp, const float* __restrict__ Uhp, const float* __restrict__ bhp)
{
    __shared__ __align__(16) float hs[WB_NE];

    const int tid  = threadIdx.x;
    const int lane = tid & 31;
    const int wave = tid >> 5;                 // 0..15
    const int row  = blockIdx.x * 16 + wave;

    for (int i = tid; i < WB_NE / 4; i += 512)
        *(v4f*)&hs[i * 4] = *(const v4f*)&h_in_b[i * 4];
    __syncthreads();

    const v4f* arow = (const v4f*)(adj_b + (size_t)row * WB_NE);
    const v4f* h4p  = (const v4f*)hs;
    float acc = 0.f;
#pragma unroll 4
    for (int i = 0; i < (WB_NE / 4) / 32; ++i) {
        const int idx = i * 32 + lane;         // coalesced 16B/lane stream
        v4f a4 = NT ? __builtin_nontemporal_load(arow + idx) : arow[idx];
        v4f h4 = h4p[idx];
        acc += a4.x * h4.x + a4.y * h4.y + a4.z * h4.z + a4.w * h4.w;
    }
    acc = wave_reduce_sum(acc);

    if (lane == 0) {
        const float h  = hs[row];
        const float a  = Wm[0] * acc + bm[0];
        const float z  = 1.f / (1.f + expf(-(a * Wzp[0] + h * Uzp[0] + bzp[0])));
        const float r  = 1.f / (1.f + expf(-(a * Wrp[0] + h * Urp[0] + brp[0])));
        const float ht = tanhf(a * Whp[0] + (r * h) * Uhp[0] + bhp[0]);
        h_out_b[row] = (1.f - z) * h + z * ht;
    }
}

// ---------------------------------------------------------------------------
// K3: stable rank-based keep of the KEEP_N smallest scores + prefix-scan
//     positions. One 1024-thread block per batch; 4 nodes per thread.
// ---------------------------------------------------------------------------
#define SEL_T 1024
__global__ __launch_bounds__(SEL_T) void select_kernel(
    const float* __restrict__ S, int* __restrict__ pos_out, float* __restrict__ mask_out)
{
    __shared__ float s[WB_NE];
    __shared__ int   wsum[SEL_T];
    const int b   = blockIdx.x;
    const int tid = threadIdx.x;

    const float* Sb = S + (size_t)b * WB_NE;
    for (int i = tid; i < WB_NE; i += SEL_T) s[i] = Sb[i];
    __syncthreads();

    int keep[4];
    int local = 0;
#pragma unroll
    for (int q = 0; q < 4; ++q) {
        const int   i  = tid * 4 + q;
        const float si = s[i];
        int cnt = 0;
        for (int j = 0; j < WB_NE; ++j) {
            const float sj = s[j];
            cnt += (sj < si) || (sj == si && j < i);   // stable ascending rank
        }
        keep[q] = (cnt < KEEP_N) ? 1 : 0;              // among KEEP_N smallest
        local += keep[q];
    }

    // inclusive Hillis-Steele scan of per-thread counts
    wsum[tid] = local;
    __syncthreads();
    for (int off = 1; off < SEL_T; off <<= 1) {
        int v = (tid >= off) ? wsum[tid - off] : 0;
        __syncthreads();
        wsum[tid] += v;
        __syncthreads();
    }
    int p = (tid == 0) ? 0 : wsum[tid - 1];            // exclusive prefix

#pragma unroll
    for (int q = 0; q < 4; ++q) {
        const int i = tid * 4 + q;
        if (keep[q]) { pos_out[(size_t)b * WB_NE + i] = p; ++p; }
        else         { pos_out[(size_t)b * WB_NE + i] = -1;    }
        mask_out[(size_t)b * WB_NE + i] = keep[q] ? 1.0f : 0.0f;
    }
}

// ---------------------------------------------------------------------------
// K4: gather kept rows of H_e into the output (original order preserved).
//     One 128-thread block per (b, node); float4 row copy.
// ---------------------------------------------------------------------------
__global__ __launch_bounds__(128) void gather_kernel(
    const float* __restrict__ He, const int* __restrict__ pos, float* __restrict__ out)
{
    const int n = blockIdx.x % WB_NE;
    const int b = blockIdx.x / WB_NE;
    const int p = pos[(size_t)b * WB_NE + n];
    if (p < 0) return;
    const v4f* src = (const v4f*)(He + ((size_t)b * WB_NE + n) * WB_H);
    v4f*       dst = (v4f*)(out + ((size_t)b * KEEP_N + p) * WB_H);
    dst[threadIdx.x] = src[threadIdx.x];
}

// ---------------------------------------------------------------------------
extern "C" void kernel_launch(void* const* d_in, const int* in_sizes, int n_in,
                              void* d_out, int out_size, void* d_ws, size_t ws_size,
                              hipStream_t stream) {
    const float* He   = (const float*)d_in[0];
    const float* Hc   = (const float*)d_in[1];
    const float* adj  = (const float*)d_in[2];
    const float* Wsc  = (const float*)d_in[3];
    const float* Wm   = (const float*)d_in[4];
    const float* bm   = (const float*)d_in[5];
    const float* Wz   = (const float*)d_in[6];
    const float* Uz   = (const float*)d_in[7];
    const float* bz   = (const float*)d_in[8];
    const float* Wr   = (const float*)d_in[9];
    const float* Ur   = (const float*)d_in[10];
    const float* br   = (const float*)d_in[11];
    const float* Wh   = (const float*)d_in[12];
    const float* Uh   = (const float*)d_in[13];
    const float* bh   = (const float*)d_in[14];

    float* out_ref  = (float*)d_out;                               // [B, KEEP_N, H]
    float* out_mask = out_ref + (size_t)WB_B * KEEP_N * WB_H;      // [B, NE] (0/1)

    // workspace layout
    float* ws_csum = (float*)d_ws;                     // B*NC
    float* h0      = ws_csum + WB_B * WB_NC;           // B*NE
    float* h1      = h0 + WB_B * WB_NE;                // B*NE
    int*   pos     = (int*)(h1 + WB_B * WB_NE);        // B*NE

    // K0: claim row sums
    csum_kernel<<<(WB_B * WB_NC) / 4, 128, 0, stream>>>(Hc, ws_csum);

    // K1: logits GEMM (WMMA f32) + softmax/relevance/score1 -> h0
    score_kernel<<<WB_B * (WB_NE / 16), 256, 0, stream>>>(He, Hc, Wsc, ws_csum, h0);

    // K2: three GGNN steps, issued PER BATCH so the 64 MB adj slab stays
    //     resident in the 192 MB L2 across that batch's three mat-vecs
    //     (steps 2-3 hit L2 instead of HBM: 1.6 GB -> 512 MB HBM traffic).
    //     Final step per batch uses non-temporal adj reads (last use).
    for (int b = 0; b < WB_B; ++b) {
        const float* adj_b = adj + (size_t)b * WB_NE * WB_NE;
        float* h0b = h0 + (size_t)b * WB_NE;
        float* h1b = h1 + (size_t)b * WB_NE;
        ggnn_step_kernel<false><<<WB_NE / 16, 512, 0, stream>>>(adj_b, h0b, h1b,
            Wm, bm, Wz, Uz, bz, Wr, Ur, br, Wh, Uh, bh);
        ggnn_step_kernel<false><<<WB_NE / 16, 512, 0, stream>>>(adj_b, h1b, h0b,
            Wm, bm, Wz, Uz, bz, Wr, Ur, br, Wh, Uh, bh);
        ggnn_step_kernel<true ><<<WB_NE / 16, 512, 0, stream>>>(adj_b, h0b, h1b,
            Wm, bm, Wz, Uz, bz, Wr, Ur, br, Wh, Uh, bh);
    }

    // K3: rank + positions + mask
    select_kernel<<<WB_B, SEL_T, 0, stream>>>(h1, pos, out_mask);

    // K4: gather kept rows
    gather_kernel<<<WB_B * WB_NE, 128, 0, stream>>>(He, pos, out_ref);
}